// layer_N_C_d_d_bilinear_attention_cov2cor_spd_37443524886536
// MI455X (gfx1250) — compile-verified
//
#include <hip/hip_runtime.h>
#include <math.h>

// ---------------------------------------------------------------------------
// N=64, C=64, d=64 fp32 bilinear attention with cov2cor + SPD softmax.
// One workgroup (128 threads = 4 wave32) per (n, c_out):
//   phase A: cor(n,ci) computed on the fly, FMA-mixed into Q/K/V (LDS)
//   phase B: T1 = K^T Q ; A_pre = T1 K ; SPD-normalize ; T2 = A V ;
//            out = inputs + T2 A^T      -- all 64x64x64 matmuls on
//            v_wmma_f32_16x16x4_f32 tiles.
// ---------------------------------------------------------------------------

typedef float v2f __attribute__((ext_vector_type(2)));
typedef float v8f __attribute__((ext_vector_type(8)));

#define DD   64
#define DD2  4096   // 64*64

// ---- WMMA fragment helpers (layouts per CDNA5 ISA 7.12.2, 32-bit, wave32) --

// A 16x4 (MxK): lanes 0-15 -> M=lane, K={k0,k0+1}; lanes 16-31 -> M=lane-16, K={k0+2,k0+3}
template <bool TRANS>
__device__ __forceinline__ v2f load_a_frag(const float* __restrict__ M,
                                           int lane, int m0, int k0) {
  int mr = m0 + (lane & 15);
  int kk = k0 + ((lane >> 4) << 1);
  v2f a;
  if (!TRANS) {
    a.x = M[mr * DD + kk];
    a.y = M[mr * DD + kk + 1];
  } else {               // A = Mat^T : A[m][k] = Mat[k][m]
    a.x = M[kk * DD + mr];
    a.y = M[(kk + 1) * DD + mr];
  }
  return a;
}

// B 4x16 (KxN): lanes 0-15 -> N=lane, K={k0,k0+1}; lanes 16-31 -> N=lane-16, K={k0+2,k0+3}
template <bool TRANS>
__device__ __forceinline__ v2f load_b_frag(const float* __restrict__ M,
                                           int lane, int k0, int n0) {
  int nn = n0 + (lane & 15);
  int kb = k0 + ((lane >> 4) << 1);
  v2f b;
  if (!TRANS) {
    b.x = M[kb * DD + nn];
    b.y = M[(kb + 1) * DD + nn];
  } else {               // B = Mat^T : B[k][n] = Mat[n][k]
    b.x = M[nn * DD + kb];
    b.y = M[nn * DD + kb + 1];
  }
  return b;
}

// C/D 16x16: VGPR r -> lanes 0-15: row m0+r, col n0+lane; lanes 16-31: row m0+r+8
__device__ __forceinline__ void store_d_frag(float* __restrict__ Dst,
                                             int lane, int m0, int n0, v8f c) {
  int col = n0 + (lane & 15);
  int rb  = m0 + ((lane >> 4) << 3);
#pragma unroll
  for (int r = 0; r < 8; ++r) Dst[(rb + r) * DD + col] = c[r];
}

// 64x64 += free: D = opA(A) * opB(B), 4 waves each own one 16-row strip.
template <bool TA, bool TB>
__device__ __forceinline__ void wg_mm64(const float* __restrict__ A,
                                        const float* __restrict__ B,
                                        float* __restrict__ Dst,
                                        int lane, int wave) {
  const int m0 = wave << 4;
#pragma unroll
  for (int nt = 0; nt < 4; ++nt) {
    const int n0 = nt << 4;
    v8f c = {0.f, 0.f, 0.f, 0.f, 0.f, 0.f, 0.f, 0.f};
#pragma unroll
    for (int kt = 0; kt < 16; ++kt) {
      v2f a = load_a_frag<TA>(A, lane, m0, kt << 2);
      v2f b = load_b_frag<TB>(B, lane, kt << 2, n0);
      c = __builtin_amdgcn_wmma_f32_16x16x4_f32(
          /*neg_a=*/false, a, /*neg_b=*/false, b,
          /*c_mod=*/(short)0, c, /*reuse_a=*/false, /*reuse_b=*/false);
    }
    store_d_frag(Dst, lane, m0, n0, c);
  }
}

// ---------------------------------------------------------------------------
// Kernel 1: inv[n,c,i] = 1/sqrt(max(|inputs[n,c,i,i]|, 1e-4))
// ---------------------------------------------------------------------------
__global__ void k_invdiag(const float* __restrict__ inp, float* __restrict__ invd) {
  const int b = blockIdx.x;         // n*C + c, 4096 blocks
  const int i = threadIdx.x;        // 0..63
  float x = inp[(size_t)b * DD2 + i * (DD + 1)];
  float d = fmaxf(fabsf(x), 1e-4f);
  invd[b * DD + i] = rsqrtf(d);
}

// ---------------------------------------------------------------------------
// Kernel 2: fused channel-mix + attention chain. grid = (C, N), block = 128.
// ---------------------------------------------------------------------------
__global__ __launch_bounds__(128)
void k_attn(const float* __restrict__ inp,
            const float* __restrict__ wq,
            const float* __restrict__ wk,
            const float* __restrict__ wv,
            const float* __restrict__ invd,
            float* __restrict__ out) {
  const int co   = blockIdx.x;      // output channel
  const int n    = blockIdx.y;      // batch
  const int tid  = threadIdx.x;     // 0..127
  const int lane = tid & 31;
  const int wave = tid >> 5;

  __shared__ __align__(16) float Qs[DD2];
  __shared__ __align__(16) float Ks[DD2];
  __shared__ __align__(16) float Vs[DD2];
  __shared__ __align__(16) float Ts[DD2];
  __shared__ float red[128];
  __shared__ float dinv[DD];

  const float* inp_n = inp + (size_t)n * DD * DD2;

  // stage invdiag for batch n (64 ch x 64) into Ts (free until first matmul)
#pragma unroll
  for (int i = 0; i < 32; ++i)
    Ts[i * 128 + tid] = invd[(size_t)n * DD2 + i * 128 + tid];
  __syncthreads();

  // ---- phase A: Q/K/V[d,d] for channel co, cor computed on the fly --------
  float4 aq[8], ak[8], av[8];
#pragma unroll
  for (int i = 0; i < 8; ++i) {
    aq[i] = make_float4(0.f, 0.f, 0.f, 0.f);
    ak[i] = make_float4(0.f, 0.f, 0.f, 0.f);
    av[i] = make_float4(0.f, 0.f, 0.f, 0.f);
  }

  for (int ci = 0; ci < DD; ++ci) {
    const float q = wq[ci * DD + co];
    const float k = wk[ci * DD + co];
    const float v = wv[ci * DD + co];
    const float* __restrict__ xm = inp_n + ci * DD2;
    const float* __restrict__ iv = &Ts[ci * DD];
    if (ci + 1 < DD) __builtin_prefetch(xm + DD2, 0, 0);  // global_prefetch_b8
#pragma unroll
    for (int i = 0; i < 8; ++i) {
      const int p  = i * 512 + tid * 4;       // 4 consecutive cols, same row
      const int r  = p >> 6;
      const int c0 = p & 63;
      float4 x  = *reinterpret_cast<const float4*>(xm + p);
      float4 ic = *reinterpret_cast<const float4*>(iv + c0);
      const float ir = iv[r];
      float cx = fminf(1.f, fmaxf(-1.f, x.x * ir * ic.x));
      float cy = fminf(1.f, fmaxf(-1.f, x.y * ir * ic.y));
      float cz = fminf(1.f, fmaxf(-1.f, x.z * ir * ic.z));
      float cw = fminf(1.f, fmaxf(-1.f, x.w * ir * ic.w));
      aq[i].x += q * cx; aq[i].y += q * cy; aq[i].z += q * cz; aq[i].w += q * cw;
      ak[i].x += k * cx; ak[i].y += k * cy; ak[i].z += k * cz; ak[i].w += k * cw;
      av[i].x += v * cx; av[i].y += v * cy; av[i].z += v * cz; av[i].w += v * cw;
    }
  }
  __syncthreads();   // done reading Ts (invdiag)
#pragma unroll
  for (int i = 0; i < 8; ++i) {
    const int p = i * 512 + tid * 4;
    *reinterpret_cast<float4*>(Qs + p) = aq[i];
    *reinterpret_cast<float4*>(Ks + p) = ak[i];
    *reinterpret_cast<float4*>(Vs + p) = av[i];
  }
  __syncthreads();

  // ---- phase B: matmul chain on WMMA --------------------------------------
  wg_mm64<true,  false>(Ks, Qs, Ts, lane, wave);   // T1 = K^T Q
  __syncthreads();
  wg_mm64<false, false>(Ts, Ks, Qs, lane, wave);   // A_pre = T1 K  (overwrites Q)
  __syncthreads();

  // ---- SPD normalize (soft_pd_max) on Qs ----------------------------------
  float mx = -INFINITY;
#pragma unroll
  for (int i = 0; i < 32; ++i) mx = fmaxf(mx, Qs[i * 128 + tid]);
  red[tid] = mx;
  __syncthreads();
  for (int s = 64; s > 0; s >>= 1) {
    if (tid < s) red[tid] = fmaxf(red[tid], red[tid + s]);
    __syncthreads();
  }
  mx = red[0];
  __syncthreads();

#pragma unroll
  for (int i = 0; i < 32; ++i) {
    const int p = i * 128 + tid;
    Qs[p] = expf(Qs[p] - mx);
  }
  __syncthreads();

  {  // row sums: 2 threads per row
    const int r = tid >> 1;
    const int h = (tid & 1) * 32;
    float s = 0.f;
#pragma unroll
    for (int j = 0; j < 32; ++j) s += Qs[r * DD + h + j];
    red[tid] = s;
  }
  __syncthreads();
  if (tid < DD) dinv[tid] = red[2 * tid] + red[2 * tid + 1];
  __syncthreads();
  if (tid < DD) red[tid] = dinv[tid];
  __syncthreads();
  for (int s = 32; s > 0; s >>= 1) {
    if (tid < s) red[tid] += red[tid + s];
    __syncthreads();
  }
  const float total = red[0];
  __syncthreads();
  if (tid < DD) {
    float dv = fmaxf(dinv[tid], total * 1e-5f);   // sum(dia)/100000
    dv = fmaxf(dv, 1e-5f);
    dinv[tid] = rsqrtf(dv);
  }
  __syncthreads();
#pragma unroll
  for (int i = 0; i < 32; ++i) {
    const int p = i * 128 + tid;
    Qs[p] *= dinv[p >> 6] * dinv[p & 63];         // A = e * inv_r * inv_c
  }
  __syncthreads();

  wg_mm64<false, false>(Qs, Vs, Ts, lane, wave);  // T2 = A V (overwrites T1)
  __syncthreads();

  // ---- AXA = T2 A^T, fused residual add, straight to HBM ------------------
  const float* __restrict__ ibase = inp + ((size_t)n * DD + co) * DD2;
  float* __restrict__       obase = out + ((size_t)n * DD + co) * DD2;
  const int m0 = wave << 4;
#pragma unroll
  for (int nt = 0; nt < 4; ++nt) {
    const int n0 = nt << 4;
    v8f c = {0.f, 0.f, 0.f, 0.f, 0.f, 0.f, 0.f, 0.f};
#pragma unroll
    for (int kt = 0; kt < 16; ++kt) {
      v2f a = load_a_frag<false>(Ts, lane, m0, kt << 2);
      v2f b = load_b_frag<true >(Qs, lane, kt << 2, n0);  // B = A^T
      c = __builtin_amdgcn_wmma_f32_16x16x4_f32(
          false, a, false, b, (short)0, c, false, false);
    }
    const int col = n0 + (lane & 15);
    const int rb  = m0 + ((lane >> 4) << 3);
#pragma unroll
    for (int r = 0; r < 8; ++r) {
      const int idx = (rb + r) * DD + col;
      obase[idx] = ibase[idx] + c[r];
    }
  }
}

// ---------------------------------------------------------------------------
extern "C" void kernel_launch(void* const* d_in, const int* in_sizes, int n_in,
                              void* d_out, int out_size, void* d_ws, size_t ws_size,
                              hipStream_t stream) {
  (void)in_sizes; (void)n_in; (void)out_size; (void)ws_size;
  const float* inputs = (const float*)d_in[0];
  const float* w_q    = (const float*)d_in[1];
  const float* w_k    = (const float*)d_in[2];
  const float* w_v    = (const float*)d_in[3];
  float* invd = (float*)d_ws;          // N*C*d floats = 1 MB scratch
  float* outp = (float*)d_out;

  k_invdiag<<<dim3(64 * 64), dim3(64), 0, stream>>>(inputs, invd);
  k_attn<<<dim3(64, 64), dim3(128), 0, stream>>>(inputs, w_q, w_k, w_v, invd, outp);
}